// NeuralGARCHLayer_82197084111503
// MI455X (gfx1250) — compile-verified
//
#include <hip/hip_runtime.h>
#include <math.h>

#define NROWS  8192
#define SLEN   2048
#define NCHUNK 16
#define TCHUNK 128

typedef __attribute__((ext_vector_type(2))) float v2f;
typedef __attribute__((ext_vector_type(8))) float v8f;

__device__ __forceinline__ void garch_coeffs(const float* om, const float* al,
                                             const float* be,
                                             float& w, float& a, float& b) {
  float o  = om[0];
  float A  = al[0];
  float Bt = be[0];
  w = log1pf(expf(o));              // softplus(omega)
  a = 0.2f / (1.0f + expf(-A));     // 0.2 * sigmoid(alpha)
  b = 0.8f / (1.0f + expf(-Bt));    // 0.8 * sigmoid(beta)
}

__device__ __forceinline__ float bpow(float b, int e) {
  float r = 1.0f;
  for (int i = 0; i < e; ++i) r *= b;
  return r;
}

// L[m][k] = b^(m-k) for k<=m else 0  (lower-triangular resolvent of the recurrence)
__device__ __forceinline__ float lent(float b, int m, int k) {
  return (k <= m) ? bpow(b, m - k) : 0.0f;
}

// ---------------------------------------------------------------------------
// Kernel 1: per (row, chunk) partial sums: S1 = sum r, S2 = sum r^2,
//           W = sum_j b^(T-1-j) * r_j^2  (Horner form, exact chunk transfer term)
// Regular (RT) loads: pulls the 64MB returns array into the 192MB L2 so
// kernel 3's re-read is an L2 hit (HBM traffic stays ~64MB read + 64MB write).
// ---------------------------------------------------------------------------
__global__ void garch_chunk_stats(const float* __restrict__ ret,
                                  const float* om, const float* al, const float* be,
                                  float* __restrict__ s1, float* __restrict__ s2,
                                  float* __restrict__ wa) {
  int idx   = blockIdx.x * blockDim.x + threadIdx.x;   // = row*NCHUNK + chunk
  int row   = idx >> 4;
  int chunk = idx & (NCHUNK - 1);
  float w, a, b;
  garch_coeffs(om, al, be, w, a, b);
  (void)w; (void)a;

  const float4* p = reinterpret_cast<const float4*>(
      ret + (size_t)row * SLEN + (size_t)chunk * TCHUNK);

  float sum = 0.0f, ssq = 0.0f, acc = 0.0f;
#pragma unroll 4
  for (int i = 0; i < TCHUNK / 4; ++i) {
    float4 v = p[i];
    float q0 = v.x * v.x, q1 = v.y * v.y, q2 = v.z * v.z, q3 = v.w * v.w;
    sum += (v.x + v.y) + (v.z + v.w);
    ssq += (q0 + q1) + (q2 + q3);
    acc = fmaf(acc, b, q0);
    acc = fmaf(acc, b, q1);
    acc = fmaf(acc, b, q2);
    acc = fmaf(acc, b, q3);
  }
  s1[idx] = sum;
  s2[idx] = ssq;
  wa[idx] = acc;
}

// ---------------------------------------------------------------------------
// Kernel 2: per row: unbiased variance -> h0; 16-step chunk-boundary scan:
//           g_{c+1} = b^T g_c + w*geo + a*W_c ; also emit out[row][0]=sqrt(h0)
// ---------------------------------------------------------------------------
__global__ void garch_row_scan(const float* __restrict__ s1,
                               const float* __restrict__ s2,
                               const float* __restrict__ wa,
                               const float* om, const float* al, const float* be,
                               float* __restrict__ g, float* __restrict__ out) {
  int row = blockIdx.x * blockDim.x + threadIdx.x;
  float w, a, b;
  garch_coeffs(om, al, be, w, a, b);

  float S1 = 0.0f, S2 = 0.0f;
#pragma unroll
  for (int c = 0; c < NCHUNK; ++c) {
    S1 += s1[row * NCHUNK + c];
    S2 += s2[row * NCHUNK + c];
  }
  float mean = S1 / (float)SLEN;
  float var  = (S2 - S1 * mean) / (float)(SLEN - 1);   // ddof=1
  float h0   = var + 1e-6f;

  float bT = b;                       // b^128 via 7 squarings
#pragma unroll
  for (int i = 0; i < 7; ++i) bT *= bT;
  float geo = (1.0f - bT) / (1.0f - b);

  out[(size_t)row * SLEN] = __builtin_amdgcn_sqrtf(h0);

  float gc = h0;
  g[row * NCHUNK + 0] = gc;
#pragma unroll
  for (int c = 0; c < NCHUNK - 1; ++c) {
    gc = fmaf(bT, gc, fmaf(a, wa[row * NCHUNK + c], w * geo));
    g[row * NCHUNK + c + 1] = gc;
  }
}

// ---------------------------------------------------------------------------
// Kernel 3: WMMA chunk expansion. One wave = 16 rows x one 128-step chunk.
// Per 16-step sub-block:  H(16 time x 16 rows) = L(16x16) * X(16x16) + p.h_prev
// realized as 4 chained V_WMMA_F32_16X16X4_F32 (K=4 slices), C preloaded with
// the rank-1 carry term. Carry h[15][n] broadcast via wave32 shuffle.
// Loads are last-use (NT hint, data already consumed once by kernel 1 and is
// L2-resident); stores are streaming (never re-read).
// ---------------------------------------------------------------------------
__global__ void garch_expand_wmma(const float* __restrict__ ret,
                                  const float* __restrict__ g,
                                  const float* om, const float* al, const float* be,
                                  float* __restrict__ out) {
  int lane   = threadIdx.x & 31;
  int waveIb = threadIdx.x >> 5;
  int chunk  = blockIdx.y;
  int rowBase = (blockIdx.x * 4 + waveIb) * 16;
  int n  = lane & 15;      // row-in-tile (B/C/D column), also A's M index
  int hh = lane >> 4;      // lane-half selects K pair / M+8 block

  float w, a, b;
  garch_coeffs(om, al, be, w, a, b);

  // A operand: L matrix, K slices s: lane holds L[m][4s+2hh] , L[m][4s+2hh+1]
  int m = n;
  v2f A0 = { lent(b, m,  0 + 2 * hh), lent(b, m,  1 + 2 * hh) };
  v2f A1 = { lent(b, m,  4 + 2 * hh), lent(b, m,  5 + 2 * hh) };
  v2f A2 = { lent(b, m,  8 + 2 * hh), lent(b, m,  9 + 2 * hh) };
  v2f A3 = { lent(b, m, 12 + 2 * hh), lent(b, m, 13 + 2 * hh) };

  // carry powers: C[v] scale = b^(v + 8*hh + 1)
  float b2 = b * b, b4 = b2 * b2, b8 = b4 * b4;
  float pcv = hh ? (b8 * b) : b;
  float pc[8];
#pragma unroll
  for (int v = 0; v < 8; ++v) { pc[v] = pcv; pcv *= b; }

  const float* rrow = ret + (size_t)(rowBase + n) * SLEN + (size_t)chunk * TCHUNK;
  const v2f*   rp   = reinterpret_cast<const v2f*>(rrow);
  float*       orow = out + (size_t)(rowBase + n) * SLEN;

  float hprev = g[(rowBase + n) * NCHUNK + chunk];   // h[chunk*T]
  int tbase0 = chunk * TCHUNK;

  for (int sb = 0; sb < 8; ++sb) {
    // X[j][n] = w + a * r[row n][chunk*T + 16*sb + j]^2 ; lane holds j = 4s+2hh,+1
    v2f r0 = __builtin_nontemporal_load(rp + 8 * sb + 0 + hh);
    v2f r1 = __builtin_nontemporal_load(rp + 8 * sb + 2 + hh);
    v2f r2 = __builtin_nontemporal_load(rp + 8 * sb + 4 + hh);
    v2f r3 = __builtin_nontemporal_load(rp + 8 * sb + 6 + hh);
    v2f X0 = { fmaf(a, r0.x * r0.x, w), fmaf(a, r0.y * r0.y, w) };
    v2f X1 = { fmaf(a, r1.x * r1.x, w), fmaf(a, r1.y * r1.y, w) };
    v2f X2 = { fmaf(a, r2.x * r2.x, w), fmaf(a, r2.y * r2.y, w) };
    v2f X3 = { fmaf(a, r3.x * r3.x, w), fmaf(a, r3.y * r3.y, w) };

    v8f C;
#pragma unroll
    for (int v = 0; v < 8; ++v) C[v] = pc[v] * hprev;   // rank-1 carry term

    C = __builtin_amdgcn_wmma_f32_16x16x4_f32(false, A0, false, X0, (short)0, C, false, false);
    C = __builtin_amdgcn_wmma_f32_16x16x4_f32(false, A1, false, X1, (short)0, C, false, false);
    C = __builtin_amdgcn_wmma_f32_16x16x4_f32(false, A2, false, X2, (short)0, C, false, false);
    C = __builtin_amdgcn_wmma_f32_16x16x4_f32(false, A3, false, X3, (short)0, C, false, false);

    // next carry: D[15][n] lives in VGPR7 of lane 16+n
    hprev = __shfl(C[7], 16 + n, 32);

    // D[i][n] = h[tbase0 + 16*sb + 1 + i]; lane writes i = v + 8*hh.
    // Single hoisted guard: only (chunk=15, sb=7, hh=1, v=7) -> t=2048 is OOB.
    int t0 = tbase0 + 16 * sb + 1 + 8 * hh;
    if (t0 + 7 < SLEN) {
#pragma unroll
      for (int v = 0; v < 8; ++v)
        __builtin_nontemporal_store(__builtin_amdgcn_sqrtf(C[v]), orow + t0 + v);
    } else {
#pragma unroll
      for (int v = 0; v < 8; ++v)
        if (t0 + v < SLEN)
          __builtin_nontemporal_store(__builtin_amdgcn_sqrtf(C[v]), orow + t0 + v);
    }
  }
}

// ---------------------------------------------------------------------------
extern "C" void kernel_launch(void* const* d_in, const int* in_sizes, int n_in,
                              void* d_out, int out_size, void* d_ws, size_t ws_size,
                              hipStream_t stream) {
  const float* ret = (const float*)d_in[0];
  const float* om  = (const float*)d_in[1];
  const float* al  = (const float*)d_in[2];
  const float* be  = (const float*)d_in[3];
  float* out = (float*)d_out;

  float* ws = (float*)d_ws;
  float* s1 = ws;
  float* s2 = s1 + NROWS * NCHUNK;
  float* wa = s2 + NROWS * NCHUNK;
  float* g  = wa + NROWS * NCHUNK;   // 4 * 131072 * 4B = 2 MB total scratch

  garch_chunk_stats<<<(NROWS * NCHUNK) / 256, 256, 0, stream>>>(ret, om, al, be, s1, s2, wa);
  garch_row_scan<<<NROWS / 256, 256, 0, stream>>>(s1, s2, wa, om, al, be, g, out);
  dim3 grid3(NROWS / 64, NCHUNK);
  garch_expand_wmma<<<grid3, 128, 0, stream>>>(ret, g, om, al, be, out);
}